// LengthRegulator_86517821210959
// MI455X (gfx1250) — compile-verified
//
#include <hip/hip_runtime.h>

// LengthRegulator for MI455X (gfx1250, wave32).
// B=32, T=1024, D=384, MAXLEN=8192.
// Three-phase: (A) per-batch scan of durations, (B) precompute searchsorted
// index map (idx or -1) once per output row, (C) lean streaming gather:
// per row 1x b32 idx load + 3x b128 loads + 3x b128 NT stores.
// Roofline ~18.5us (384MB NT stores + 48MB L2-resident x reads @ 23.3 TB/s).
// No matrix math => no WMMA; target mix is b128 vmem + ds.

#define B_      32
#define T_      1024
#define D_      384
#define MAXLEN_ 8192
#define D4_     (D_ / 4)          // 96 float4 per row

typedef float __attribute__((ext_vector_type(4))) v4f;
typedef int   __attribute__((ext_vector_type(4))) v4i;

// ---------------------------------------------------------------------------
// Kernel A: per-batch inclusive scan of duration (clamped at 0).
// One 1024-thread block (32 waves) per batch row. Two-level wave32 shfl scan.
// Writes cum[b,t] to workspace and total[b] (as float) into d_out tail.
// ---------------------------------------------------------------------------
__global__ __launch_bounds__(T_) void lr_scan_kernel(
    const int* __restrict__ dur,       // [B, T]
    int*       __restrict__ cum,       // [B, T] workspace
    float*     __restrict__ total_out) // [B] -> d_out + B*MAXLEN*D
{
    __shared__ int wsum[32];
    const int b    = blockIdx.x;
    const int t    = threadIdx.x;
    const int lane = t & 31;           // wave32
    const int wave = t >> 5;

    int d = dur[b * T_ + t];
    int v = d > 0 ? d : 0;             // jnp.maximum(duration, 0)

    // Inclusive scan within the wave.
    #pragma unroll
    for (int off = 1; off < 32; off <<= 1) {
        const int n = __shfl_up(v, off, 32);
        if (lane >= off) v += n;
    }
    if (lane == 31) wsum[wave] = v;
    __syncthreads();

    // Scan of the 32 per-wave sums by wave 0.
    if (wave == 0) {
        int s = wsum[lane];
        #pragma unroll
        for (int off = 1; off < 32; off <<= 1) {
            const int n = __shfl_up(s, off, 32);
            if (lane >= off) s += n;
        }
        wsum[lane] = s;
    }
    __syncthreads();

    if (wave > 0) v += wsum[wave - 1];

    cum[b * T_ + t] = v;
    if (t == T_ - 1) total_out[b] = (float)v;
}

// ---------------------------------------------------------------------------
// Kernel B: precompute idxmap[row] = searchsorted(cum[b], pos, 'right')
// clamped to T-1, or -1 if pos >= total. One thread per output row; 256 rows
// per block all share one b (8192 % 256 == 0), cum row staged via b128 LDS.
// ---------------------------------------------------------------------------
__global__ __launch_bounds__(256) void lr_search_kernel(
    const int* __restrict__ cum,       // [B, T]
    int*       __restrict__ idxmap)    // [B * MAXLEN]
{
    __shared__ int scum[T_];
    const int tid  = threadIdx.x;
    const int row0 = blockIdx.x * 256;           // < 262144, fits in int
    const int b    = row0 >> 13;                 // row0 / MAXLEN_

    // 256 threads x int4 = 1024 ints: single b128 global load + b128 ds store.
    ((v4i*)scum)[tid] = ((const v4i*)(cum + b * T_))[tid];
    __syncthreads();

    const int total = scum[T_ - 1];
    const int pos   = (row0 + tid) & (MAXLEN_ - 1);

    int res = -1;
    if (pos < total) {
        // Branchless power-of-two binary search: count of cum[i] <= pos.
        int idx = 0;
        #pragma unroll
        for (int sh = 9; sh >= 0; --sh) {
            const int probe = idx + (1 << sh);
            if (scum[probe - 1] <= pos) idx = probe;
        }
        res = idx < (T_ - 1) ? idx : (T_ - 1);   // min(idx, T-1)
    }
    idxmap[row0 + tid] = res;
}

// ---------------------------------------------------------------------------
// Kernel C: streaming gather. One wave per 2 output rows (ILP=2), 8 waves =
// 16 rows per block. Per row: idx load (broadcast), 3x b128 loads from the
// L2-resident x row, 3x b128 non-temporal stores (don't evict x from L2).
// idx < 0 => zeros. No LDS, no barriers on the 384MB-store critical path.
// ---------------------------------------------------------------------------
#define ROWS_PER_WAVE 2
#define WAVES_PB      8
#define ROWS_PB       (ROWS_PER_WAVE * WAVES_PB)   // 16

__global__ __launch_bounds__(256) void lr_gather_kernel(
    const float* __restrict__ x,       // [B, T, D]
    const int*   __restrict__ idxmap,  // [B * MAXLEN]
    float*       __restrict__ out)     // [B, MAXLEN, D]
{
    const int tid  = threadIdx.x;
    const int wave = tid >> 5;                     // wave32
    const int lane = tid & 31;

    const long long rowBase = (long long)blockIdx.x * ROWS_PB;
    const int       b       = (int)(rowBase >> 13);          // uniform/block
    const long long r0      = rowBase + wave * ROWS_PER_WAVE;

    const int idx0 = idxmap[r0];
    const int idx1 = idxmap[r0 + 1];

    const v4f* __restrict__ xb  = (const v4f*)x + (long long)b * T_ * D4_;
    v4f*       __restrict__ dst = (v4f*)out + r0 * (long long)D4_;

    v4f a0 = {}, a1 = {}, a2 = {};
    v4f c0 = {}, c1 = {}, c2 = {};

    if (idx0 >= 0) {                               // uniform per wave
        const v4f* __restrict__ s = xb + (long long)idx0 * D4_;
        a0 = s[lane]; a1 = s[lane + 32]; a2 = s[lane + 64];
    }
    if (idx1 >= 0) {
        const v4f* __restrict__ s = xb + (long long)idx1 * D4_;
        c0 = s[lane]; c1 = s[lane + 32]; c2 = s[lane + 64];
    }

    __builtin_nontemporal_store(a0, dst + lane);
    __builtin_nontemporal_store(a1, dst + lane + 32);
    __builtin_nontemporal_store(a2, dst + lane + 64);
    __builtin_nontemporal_store(c0, dst + D4_ + lane);
    __builtin_nontemporal_store(c1, dst + D4_ + lane + 32);
    __builtin_nontemporal_store(c2, dst + D4_ + lane + 64);
}

// ---------------------------------------------------------------------------
// Launcher. d_in[0]=x f32 [B,T,D], d_in[1]=duration i32 [B,T], d_in[2]=max_len.
// d_out = out f32 [B,MAXLEN,D] ++ total [B] (written as float).
// d_ws: cum i32 [B,T] (128 KB) ++ idxmap i32 [B*MAXLEN] (1 MB).
// ---------------------------------------------------------------------------
extern "C" void kernel_launch(void* const* d_in, const int* in_sizes, int n_in,
                              void* d_out, int out_size, void* d_ws, size_t ws_size,
                              hipStream_t stream) {
    const float* x   = (const float*)d_in[0];
    const int*   dur = (const int*)d_in[1];
    (void)in_sizes; (void)n_in; (void)out_size; (void)ws_size;

    float* out       = (float*)d_out;
    float* total_out = out + (size_t)B_ * MAXLEN_ * D_;
    int*   cum       = (int*)d_ws;
    int*   idxmap    = cum + B_ * T_;

    lr_scan_kernel<<<B_, T_, 0, stream>>>(dur, cum, total_out);

    lr_search_kernel<<<(B_ * MAXLEN_) / 256, 256, 0, stream>>>(cum, idxmap);

    lr_gather_kernel<<<(B_ * MAXLEN_) / ROWS_PB, 256, 0, stream>>>(x, idxmap, out);
}